// SpConvModel_46918222741767
// MI455X (gfx1250) — compile-verified
//
#include <hip/hip_runtime.h>

typedef __attribute__((ext_vector_type(2))) float v2f;
typedef __attribute__((ext_vector_type(8))) float v8f;

#define BATCH 2
#define DGRID 128

// ---------------- zero workspace (dense grid) ----------------
__global__ void zero_f4_kernel(float4* __restrict__ p, long n4) {
    long i = (long)blockIdx.x * blockDim.x + threadIdx.x;
    long stride = (long)gridDim.x * blockDim.x;
    for (; i < n4; i += stride) p[i] = make_float4(0.f, 0.f, 0.f, 0.f);
}

// ---------------- scatter voxels into dense grid ----------------
__global__ void scatter_kernel(const int* __restrict__ coords,
                               const float* __restrict__ voxels,
                               float* __restrict__ dense, int n) {
    int i = blockIdx.x * blockDim.x + threadIdx.x;
    if (i >= n) return;
    int b = coords[4*i+0], z = coords[4*i+1], y = coords[4*i+2], x = coords[4*i+3];
    long base = ((((long)b * DGRID + z) * DGRID + y) * DGRID + x) * 3;
    atomicAdd(&dense[base+0], voxels[3*i+0]);
    atomicAdd(&dense[base+1], voxels[3*i+1]);
    atomicAdd(&dense[base+2], voxels[3*i+2]);
}

// compile-time im2col offset: patch index for (k, m=0), add m*CIN for position m
__host__ __device__ constexpr int im2col_off(int k, int CIN_, int KX_, int KY_, int XT_) {
    int ci = k % CIN_; int t = k / CIN_;
    int dx = t % KX_;  t /= KX_;
    int dy = t % KY_;  int dz = t / KY_;
    return ((dz * KY_ + dy) * XT_ + dx) * CIN_ + ci;
}

// ---------------- WMMA im2col valid 3D conv ----------------
// One wave per block. Tile: M = 16 output x-positions at fixed (b,z,y).
// GEMM [16 x KPAD] x [KPAD x 16] via V_WMMA_F32_16X16X4_F32 (N padded to 16).
// A/B panels pair-interleaved so each lane's fragment is ONE ds_load_b64 at a
// compile-time immediate offset. Epilogue stages C through LDS so all global
// stores are dense and contiguous.
template <int KZ, int KY, int KX, int CIN, int COUT, bool BIAS_RELU>
__global__ __launch_bounds__(32)
void conv3d_wmma_kernel(const float* __restrict__ in, const float* __restrict__ w,
                        const float* __restrict__ bias, float* __restrict__ out,
                        int Zout, int Yout, int Xout) {
    constexpr int K    = KZ * KY * KX * CIN;
    constexpr int KPAD = (K + 3) & ~3;
    constexpr int KP2  = KPAD / 2;
    constexpr int XT   = 16 + KX - 1;        // input x-extent of the tile patch
    constexpr int PN   = KZ * KY * XT * CIN;

    const int Zin = Zout + KZ - 1, Yin = Yout + KY - 1, Xin = Xout + KX - 1;

    __shared__ float sPatch[PN];             // raw input patch (x-guarded)
    __shared__ v2f   sA[KP2 * 16];           // im2col A panel, pair-interleaved
    __shared__ v2f   sB[KP2 * 16];           // weight B panel, pair-interleaved
    __shared__ float sC[16 * 17];            // C tile staging (stride 17 vs bank conflicts)
    float* sAf = (float*)sA;
    float* sBf = (float*)sB;

    const int lane = threadIdx.x;
    const int half = lane >> 4;              // 0: lanes 0-15, 1: lanes 16-31
    const int nlo  = lane & 15;              // M (for A) / N (for B/C/D)
    const int x0   = blockIdx.x * 16;
    const int y    = blockIdx.y;
    const int bz   = blockIdx.z;
    const int b    = bz / Zout;
    const int z    = bz % Zout;

    // stage zero-padded, pair-interleaved weight panel
    for (int i = lane; i < KPAD * 16; i += 32) {
        int k = i >> 4, n = i & 15;
        float v = (k < K && n < COUT) ? w[k * COUT + n] : 0.f;
        sBf[(((k >> 1) * 16) + n) * 2 + (k & 1)] = v;
    }
    // stage input patch (z,y always in-bounds by grid construction; guard x)
    for (int i = lane; i < PN; i += 32) {
        int t = i;
        int ci = t % CIN; t /= CIN;
        int lx = t % XT;  t /= XT;
        int dy = t % KY;
        int dz = t / KY;
        int ix = x0 + lx;
        float v = 0.f;
        if (ix < Xin) {
            long idx = ((((long)b * Zin + (z + dz)) * Yin + (y + dy)) * (long)Xin + ix) * CIN + ci;
            v = in[idx];
        }
        sPatch[i] = v;
    }
    __syncthreads();

    // expand im2col into the interleaved A panel; offsets are compile-time
    // constants after unrolling (lane covers (k = 2*kp + half, m = nlo))
#pragma unroll
    for (int kp = 0; kp < KP2; ++kp) {
        const int k0e = 2 * kp, k1e = 2 * kp + 1;
        int off   = half ? im2col_off(k1e < K ? k1e : 0, CIN, KX, KY, XT)
                         : im2col_off(k0e,               CIN, KX, KY, XT);
        bool ok   = half ? (k1e < K) : true;   // only the very last k can pad
        float v   = ok ? sPatch[off + nlo * CIN] : 0.f;
        sAf[(kp * 16 + nlo) * 2 + half] = v;
    }
    __syncthreads();

    // branch-free WMMA loop: per step, ONE ds_load_b64 for A, ONE for B,
    // both at immediate offsets (k0*64 bytes) from a fixed per-lane base.
    const int fbase = half * 16 + nlo;
    v8f acc = {};
#pragma unroll
    for (int k0 = 0; k0 < KPAD; k0 += 4) {
        v2f afrag = sA[(k0 >> 1) * 16 + fbase];
        v2f bfrag = sB[(k0 >> 1) * 16 + fbase];
        acc = __builtin_amdgcn_wmma_f32_16x16x4_f32(
            /*neg_a=*/false, afrag, /*neg_b=*/false, bfrag,
            /*c_mod=*/(short)0, acc, /*reuse_a=*/false, /*reuse_b=*/false);
    }

    // C/D layout: VGPR r -> M = r (lanes 0-15) or r+8 (lanes 16-31), N = lane&15.
    // Stage through LDS, then emit dense contiguous stores.
#pragma unroll
    for (int r = 0; r < 8; ++r) {
        sC[(r + half * 8) * 17 + nlo] = acc[r];
    }
    __syncthreads();

    const int  xs    = (Xout - x0) < 16 ? (Xout - x0) : 16;  // valid x positions
    const int  valid = xs * COUT;
    const long rbase = ((((long)b * Zout + z) * Yout + y) * (long)Xout + x0) * COUT;
    for (int i = lane; i < valid; i += 32) {
        int m = i / COUT;
        int n = i - m * COUT;
        float v = sC[m * 17 + n];
        if (BIAS_RELU) { v += bias[n]; v = v > 0.f ? v : 0.f; }
        out[rbase + i] = v;
    }
}

static inline size_t align256(size_t x) { return (x + 255) & ~(size_t)255; }

extern "C" void kernel_launch(void* const* d_in, const int* in_sizes, int n_in,
                              void* d_out, int out_size, void* d_ws, size_t ws_size,
                              hipStream_t stream) {
    const int*   coords = (const int*)d_in[0];
    const float* voxels = (const float*)d_in[1];
    const float* W1     = (const float*)d_in[2];
    const float* W2     = (const float*)d_in[3];
    const float* W3     = (const float*)d_in[4];
    const float* b3     = (const float*)d_in[5];
    const int N = in_sizes[0] / 4;

    // workspace layout
    char* ws = (char*)d_ws;
    const size_t denseElems = (size_t)BATCH * DGRID * DGRID * DGRID * 3;  // 12.58M
    float* dense = (float*)ws;  ws += align256(denseElems * sizeof(float));
    const size_t x1Elems = (size_t)BATCH * 127 * 127 * 127 * 9;           // 36.87M
    float* x1 = (float*)ws;     ws += align256(x1Elems * sizeof(float));
    float* x2 = (float*)ws;                                               // 19.53M

    // 1) zero the dense grid (must happen every call)
    {
        long n4 = (long)(denseElems / 4);
        int blocks = (int)((n4 + 255) / 256);
        zero_f4_kernel<<<blocks, 256, 0, stream>>>((float4*)dense, n4);
    }
    // 2) scatter-add voxels
    scatter_kernel<<<(N + 255) / 256, 256, 0, stream>>>(coords, voxels, dense, N);

    // 3) conv1: [2,128^3,3] -> [2,127^3,9], W1 = [2,2,2,3,9]
    conv3d_wmma_kernel<2, 2, 2, 3, 9, false>
        <<<dim3((127 + 15) / 16, 127, BATCH * 127), 32, 0, stream>>>(
            dense, W1, nullptr, x1, 127, 127, 127);

    // 4) conv2: -> [2,125^3,5], W2 = [3,3,3,9,5]
    conv3d_wmma_kernel<3, 3, 3, 9, 5, false>
        <<<dim3((125 + 15) / 16, 125, BATCH * 125), 32, 0, stream>>>(
            x1, W2, nullptr, x2, 125, 125, 125);

    // 5) conv3 + bias + relu: -> [2,123^3,3] into d_out, W3 = [3,3,3,5,3]
    conv3d_wmma_kernel<3, 3, 3, 5, 3, true>
        <<<dim3((123 + 15) / 16, 123, BATCH * 123), 32, 0, stream>>>(
            x2, W3, b3, (float*)d_out, 123, 123, 123);
}